// Separator_15496242004436
// MI455X (gfx1250) — compile-verified
//
#include <hip/hip_runtime.h>
#include <math.h>

// ---------------- problem constants ----------------
#define N_FFT    4096
#define N_HOP    1024
#define WIN_T    300      // WIENER_WIN
#define N_WIN    4        // B * (T_FRAMES/WIENER_WIN)
#define BATCH    2
#define T_FRAMES 600
#define N_BINS   2049
#define N_CH     2
#define N_SRC    4
#define AUDIO_LEN 613376          // (600-1)*1024
#define N_FRAMES  9600            // B*S*C*T = 2*4*2*600
#define GEMM_M    9600            // frames
#define GEMM_N    4096            // output samples per frame
#define GEMM_K    4112            // 2*2049 (re/im interleaved) padded to 16
#define TWO_PI_OVER_N 0.001533980787885641f  // 2*pi/4096

typedef __attribute__((ext_vector_type(2))) float v2f;
typedef __attribute__((ext_vector_type(8))) float v8f;
typedef __attribute__((ext_vector_type(4))) unsigned int u32x4;
typedef __attribute__((ext_vector_type(8))) int i32x8;
typedef __attribute__((ext_vector_type(4))) int i32x4;

// ---------------- 1. init: zero max slots + Yc pad columns ----------------
__global__ void init_kernel(float* __restrict__ Ycf, int* __restrict__ maxbuf) {
  int idx = blockIdx.x * blockDim.x + threadIdx.x;
  if (idx < N_WIN) maxbuf[idx] = 0;                     // float 0.0 bit pattern
  int total = N_FRAMES * (GEMM_K - 2 * N_BINS);         // 9600*14
  if (idx < total) {
    int fr = idx / 14;
    int c  = idx % 14;
    Ycf[(size_t)fr * GEMM_K + 2 * N_BINS + c] = 0.0f;
  }
}

// ---------------- 2. generate windowed-irfft matrix G (K x N, row major) ---
// out[n] = (1/N)*win(n)*[ Yre[0] + 2*sum_{k=1..2047}(Yre[k]cos - Yim[k]sin) + Yre[2048]*(-1)^n ]
__global__ void gen_irfft_kernel(float* __restrict__ G) {
  int idx = blockIdx.x * blockDim.x + threadIdx.x;   // exactly 4112*4096 threads
  int kk = idx / GEMM_N;
  int n  = idx % GEMM_N;
  float val = 0.0f;
  if (kk < 2 * N_BINS) {
    int k = kk >> 1;
    int ph = (n * k) & (N_FFT - 1);                  // exact integer phase
    float ang = (float)ph * TWO_PI_OVER_N;
    float s, c;
    sincosf(ang, &s, &c);
    float wn = 0.5f - 0.5f * cosf((float)n * TWO_PI_OVER_N);  // hann window
    float wk = (k == 0 || k == N_FFT / 2) ? 1.0f : 2.0f;
    float scale = wn * wk * (1.0f / (float)N_FFT);
    val = (kk & 1) ? (-s * scale) : (c * scale);
  }
  G[idx] = val;
}

// ---------------- 3. per-window max |mix|^2 reduction ----------------------
__global__ __launch_bounds__(256) void maxabs_kernel(const float* __restrict__ mix,
                                                     int* __restrict__ maxbuf) {
  __shared__ float red[256];
  int win = blockIdx.x >> 7;                 // 4 windows x 128 blocks
  int blk = blockIdx.x & 127;
  int b = win >> 1, t0 = (win & 1) * WIN_T;
  const float* base = mix + (size_t)(b * T_FRAMES + t0) * N_BINS * (N_CH * 2);
  const int nElem = WIN_T * N_BINS * N_CH;   // complex count in window
  float m = 0.0f;
  for (int i = blk * 256 + threadIdx.x; i < nElem; i += 128 * 256) {
    float re = base[2 * i], im = base[2 * i + 1];
    m = fmaxf(m, re * re + im * im);
  }
  red[threadIdx.x] = m;
  __syncthreads();
  for (int s = 128; s > 0; s >>= 1) {
    if (threadIdx.x < s) red[threadIdx.x] = fmaxf(red[threadIdx.x], red[threadIdx.x + s]);
    __syncthreads();
  }
  if (threadIdx.x == 0) atomicMax(maxbuf + win, __float_as_int(red[0]));
}

__device__ __forceinline__ float max_abs_of(const int* maxbuf, int win) {
  float mv = __int_as_float(maxbuf[win]);
  return fmaxf(1.0f, sqrtf(mv) * 0.1f);     // SCALE_FACTOR = 10
}

// ---------------- 4. accumulate R[f,s] (2x2 hermitian) over t --------------
__global__ void wiener_R_kernel(const float* __restrict__ spec,
                                const float* __restrict__ mix,
                                const int* __restrict__ maxbuf,
                                float* __restrict__ Rbuf) {
  int idx = blockIdx.x * blockDim.x + threadIdx.x;
  if (idx >= N_WIN * N_BINS * N_SRC) return;
  int s   = idx & 3;
  int f   = (idx >> 2) % N_BINS;
  int win = idx / (N_BINS * N_SRC);
  int b = win >> 1, t0 = (win & 1) * WIN_T;
  float r00 = 0.f, r11 = 0.f, r01r = 0.f, r01i = 0.f, sv = 0.f;
  for (int t = 0; t < WIN_T; ++t) {
    size_t base = ((size_t)(b * T_FRAMES + t0 + t) * N_BINS + f);
    float s0 = spec[base * 8 + s];
    float s1 = spec[base * 8 + 4 + s];
    float m0r = mix[base * 4 + 0], m0i = mix[base * 4 + 1];
    float m1r = mix[base * 4 + 2], m1i = mix[base * 4 + 3];
    float n0 = m0r * m0r + m0i * m0i;
    float n1 = m1r * m1r + m1i * m1i;
    float u0r = 1.f, u0i = 0.f, u1r = 1.f, u1i = 0.f;
    if (n0 > 0.f) { float r = rsqrtf(n0); u0r = m0r * r; u0i = m0i * r; }
    if (n1 > 0.f) { float r = rsqrtf(n1); u1r = m1r * r; u1i = m1i * r; }
    float cr = u0r * u1r + u0i * u1i;        // u0 * conj(u1)
    float ci = u0i * u1r - u0r * u1i;
    float p = s0 * s1;
    r00 += s0 * s0;  r11 += s1 * s1;
    r01r += p * cr;  r01i += p * ci;
    sv  += s0 * s0 + s1 * s1;
  }
  float ma = max_abs_of(maxbuf, win);
  float inv = 1.0f / (1e-10f * ma * ma + 0.5f * sv);
  float* R = Rbuf + (size_t)idx * 4;
  R[0] = r00 * inv;  R[1] = r11 * inv;  R[2] = r01r * inv;  R[3] = r01i * inv;
}

// ---------------- 5. per-(t,f) Wiener solve -> Yc (frame-major GEMM input) -
__global__ void wiener_apply_kernel(const float* __restrict__ spec,
                                    const float* __restrict__ mix,
                                    const float* __restrict__ Rbuf,
                                    const int* __restrict__ maxbuf,
                                    float* __restrict__ Ycf) {
  int idx = blockIdx.x * blockDim.x + threadIdx.x;
  if (idx >= N_WIN * WIN_T * N_BINS) return;
  int f   = idx % N_BINS;
  int tw  = idx / N_BINS;
  int t   = tw % WIN_T;
  int win = tw / WIN_T;
  int b = win >> 1;
  int tg = (win & 1) * WIN_T + t;
  float ma = max_abs_of(maxbuf, win);
  float inv_ma = 1.0f / ma;
  size_t base = ((size_t)(b * T_FRAMES + tg) * N_BINS + f);
  const float* sp = spec + base * 8;
  const float* mx = mix + base * 4;
  float x0r = mx[0] * inv_ma, x0i = mx[1] * inv_ma;
  float x1r = mx[2] * inv_ma, x1i = mx[3] * inv_ma;
  const float* R = Rbuf + ((size_t)win * N_BINS + f) * 16;
  float c00 = 1e-5f, c11 = 1e-5f, c01r = 0.f, c01i = 0.f;   // sqrt(eps)*I reg
  float vs[N_SRC];
  float iv2 = inv_ma * inv_ma;
#pragma unroll
  for (int s = 0; s < N_SRC; ++s) {
    float s0 = sp[s], s1 = sp[4 + s];
    float v = 0.5f * (s0 * s0 + s1 * s1) * iv2;
    vs[s] = v;
    c00 += v * R[s * 4 + 0];
    c11 += v * R[s * 4 + 1];
    c01r += v * R[s * 4 + 2];
    c01i += v * R[s * 4 + 3];
  }
  float invdet = 1.0f / (c00 * c11 - (c01r * c01r + c01i * c01i));
  float ix0r = (c11 * x0r - (c01r * x1r - c01i * x1i)) * invdet;
  float ix0i = (c11 * x0i - (c01r * x1i + c01i * x1r)) * invdet;
  float ix1r = (c00 * x1r - (c01r * x0r + c01i * x0i)) * invdet;
  float ix1i = (c00 * x1i - (c01r * x0i - c01i * x0r)) * invdet;
#pragma unroll
  for (int s = 0; s < N_SRC; ++s) {
    float R00 = R[s * 4 + 0], R11 = R[s * 4 + 1];
    float pr = R[s * 4 + 2], pi = R[s * 4 + 3];
    float g0r = R00 * ix0r + (pr * ix1r - pi * ix1i);
    float g0i = R00 * ix0i + (pr * ix1i + pi * ix1r);
    float g1r = (pr * ix0r + pi * ix0i) + R11 * ix1r;
    float g1i = (pr * ix0i - pi * ix0r) + R11 * ix1i;
    float sc = vs[s] * ma;
    size_t fr0 = (size_t)(((b * N_SRC + s) * N_CH + 0) * T_FRAMES + tg);
    size_t fr1 = (size_t)(((b * N_SRC + s) * N_CH + 1) * T_FRAMES + tg);
    Ycf[fr0 * GEMM_K + 2 * f]     = sc * g0r;
    Ycf[fr0 * GEMM_K + 2 * f + 1] = sc * g0i;
    Ycf[fr1 * GEMM_K + 2 * f]     = sc * g1r;
    Ycf[fr1 * GEMM_K + 2 * f + 1] = sc * g1i;
  }
}

// ---------------- TDM 2D tile load: global -> LDS (D# per ISA ch.8) --------
__device__ __forceinline__ void tdm_load_2d(unsigned lds_off, const void* gptr,
                                            int tile_x, int tile_y,
                                            long long row_stride_elts,
                                            int pad_interval_code, int pad_amount_code,
                                            int pad_enable) {
  unsigned long long ga = (unsigned long long)gptr & ((1ULL << 57) - 1ULL);
  u32x4 g0;
  g0.x = 1u;                                           // count=1 user descriptor
  g0.y = lds_off;                                      // lds_addr
  g0.z = (unsigned)(ga & 0xFFFFFFFFu);                 // global_addr lo
  g0.w = (unsigned)((ga >> 32) & 0x1FFFFFFu) | (2u << 30);  // global_addr hi | type=2
  i32x8 g1;
  unsigned w0 = (2u << 16)                             // data_size = 4 bytes
              | ((unsigned)(pad_enable & 1) << 20)
              | ((unsigned)pad_interval_code << 22)
              | ((unsigned)pad_amount_code << 25);
  unsigned td0 = 0x40000000u, td1 = 0x40000000u;       // huge dims: no OOB clamp
  unsigned long long st0 = (unsigned long long)row_stride_elts;  // 48-bit
  g1[0] = (int)w0;
  g1[1] = (int)((td0 & 0xFFFFu) << 16);                // barrier_addr=0 | td0[15:0]
  g1[2] = (int)(((td0 >> 16) & 0xFFFFu) | ((td1 & 0xFFFFu) << 16));
  g1[3] = (int)(((td1 >> 16) & 0xFFFFu) | (((unsigned)tile_x & 0xFFFFu) << 16));
  g1[4] = (int)((unsigned)tile_y & 0xFFFFu);           // tile_dim1 | tile_dim2=0
  g1[5] = (int)(st0 & 0xFFFFFFFFu);                    // tensor_dim0_stride lo
  g1[6] = (int)((st0 >> 32) & 0xFFFFu);                // stride hi | dim1_stride lo=0
  g1[7] = 0;
  i32x4 gz = {0, 0, 0, 0};
#if __clang_major__ >= 23
  i32x8 gz8 = {0, 0, 0, 0, 0, 0, 0, 0};
  __builtin_amdgcn_tensor_load_to_lds(g0, g1, gz, gz, gz8, 0);
#else
  __builtin_amdgcn_tensor_load_to_lds(g0, g1, gz, gz, 0);
#endif
}

// ---------------- 6. fp32 WMMA GEMM: Out(9600x4096) = Yc(9600x4112)*G ------
// TDM double-buffered tiles: A 64x16 (padded to stride 20 via TDM pad),
// B 16x64 packed stride 64. One wave issues DMA, TENSORcnt-synced.
#define LDA_S 20
#define LDB_S 64
__global__ __launch_bounds__(128) void wmma_gemm_kernel(const float* __restrict__ A,   // M x K
                                                        const float* __restrict__ B,   // K x N
                                                        float* __restrict__ C) {       // M x N
  __shared__ float As[2][64 * LDA_S];
  __shared__ float Bs[2][16 * LDB_S];
  const int bm = blockIdx.x % (GEMM_M / 64);
  const int bn = blockIdx.x / (GEMM_M / 64);
  const int m0 = bm * 64;
  const int n0 = bn * 64;
  const int tid  = threadIdx.x;
  const int wave = tid >> 5;
  const int lane = tid & 31;
  const int lhi  = lane >> 4;
  const int llo  = lane & 15;

  const unsigned as_lds[2] = {(unsigned)(size_t)&As[0][0], (unsigned)(size_t)&As[1][0]};
  const unsigned bs_lds[2] = {(unsigned)(size_t)&Bs[0][0], (unsigned)(size_t)&Bs[1][0]};

  v8f acc[4];
#pragma unroll
  for (int j = 0; j < 4; ++j)
#pragma unroll
    for (int r = 0; r < 8; ++r) acc[j][r] = 0.0f;

  if (wave == 0) {   // prologue: DMA k-chunk 0 into buffer 0
    tdm_load_2d(as_lds[0], A + (size_t)m0 * GEMM_K, 16, 64, GEMM_K, 3, 3, 1);
    tdm_load_2d(bs_lds[0], B + (size_t)0 * GEMM_N + n0, 64, 16, GEMM_N, 0, 0, 0);
  }

  int cur = 0;
  for (int kc = 0; kc < GEMM_K; kc += 16, cur ^= 1) {
    __syncthreads();              // everyone done reading buffer cur^1
    if (wave == 0) {
      if (kc + 16 < GEMM_K) {     // prefetch next chunk into other buffer
        tdm_load_2d(as_lds[cur ^ 1], A + (size_t)m0 * GEMM_K + kc + 16, 16, 64, GEMM_K, 3, 3, 1);
        tdm_load_2d(bs_lds[cur ^ 1], B + (size_t)(kc + 16) * GEMM_N + n0, 64, 16, GEMM_N, 0, 0, 0);
        // in-order TENSORcnt: <=2 outstanding means current pair has landed
        __builtin_amdgcn_s_wait_tensorcnt(2);
      } else {
        __builtin_amdgcn_s_wait_tensorcnt(0);
      }
    }
    __syncthreads();              // current buffer visible to all waves
    const float* as = &As[cur][0];
    const float* bs = &Bs[cur][0];
#pragma unroll
    for (int kk = 0; kk < 4; ++kk) {
      const int k0 = kk * 4;
      // A frag (16x4 f32): lanes 0-15 -> K0,K1 ; lanes 16-31 -> K2,K3
      const float* ap = as + (wave * 16 + llo) * LDA_S + k0 + 2 * lhi;
      v2f a; a.x = ap[0]; a.y = ap[1];
#pragma unroll
      for (int j = 0; j < 4; ++j) {
        // B frag (4x16 f32): lanes 0-15 -> rows K0,K1 ; lanes 16-31 -> K2,K3
        const float* bp = bs + (k0 + 2 * lhi) * LDB_S + j * 16 + llo;
        v2f bfrag; bfrag.x = bp[0]; bfrag.y = bp[LDB_S];
        acc[j] = __builtin_amdgcn_wmma_f32_16x16x4_f32(
            false, a, false, bfrag, (short)0, acc[j], false, false);
      }
    }
  }
  const int mrow = m0 + wave * 16 + lhi * 8;   // C/D: VGPR r -> rows (r, r+8)
#pragma unroll
  for (int j = 0; j < 4; ++j)
#pragma unroll
    for (int r = 0; r < 8; ++r)
      C[(size_t)(mrow + r) * GEMM_N + n0 + j * 16 + llo] = acc[j][r];
}

// ---------------- 7. overlap-add + wsum normalize + crop -------------------
__global__ void ola_kernel(const float* __restrict__ Cmat, float* __restrict__ out) {
  int idx = blockIdx.x * blockDim.x + threadIdx.x;
  if (idx >= 16 * AUDIO_LEN) return;               // B*S*C = 16 streams
  int g = idx / AUDIO_LEN;
  int o = idx % AUDIO_LEN;
  int p = o + N_FFT / 2;                           // crop start = n_fft/2
  int tlo = (p > N_FFT - 1) ? ((p - (N_FFT - 1) + (N_HOP - 1)) >> 10) : 0;
  int thi = p >> 10; if (thi > T_FRAMES - 1) thi = T_FRAMES - 1;
  float acc = 0.0f, wsum = 0.0f;
  for (int t = tlo; t <= thi; ++t) {
    int n = p - (t << 10);
    float w = 0.5f - 0.5f * cosf((float)n * TWO_PI_OVER_N);
    wsum += w * w;
    acc += Cmat[(size_t)(g * T_FRAMES + t) * GEMM_N + n];   // window folded into G
  }
  out[idx] = acc / (wsum > 1e-11f ? wsum : 1.0f);
}

// ---------------- host launcher -------------------------------------------
extern "C" void kernel_launch(void* const* d_in, const int* in_sizes, int n_in,
                              void* d_out, int out_size, void* d_ws, size_t ws_size,
                              hipStream_t stream) {
  const float* spec = (const float*)d_in[0];   // (2,600,2049,2,4)
  const float* mix  = (const float*)d_in[1];   // (2,600,2049,2,2)
  float* out = (float*)d_out;                  // (2,4,2,613376)
  float* ws  = (float*)d_ws;

  float* Ycf  = ws;                                        // 9600*4112
  float* G    = Ycf + (size_t)GEMM_M * GEMM_K;             // 4112*4096
  float* Cmat = G + (size_t)GEMM_K * GEMM_N;               // 9600*4096
  float* Rbuf = Cmat + (size_t)GEMM_M * GEMM_N;            // 4*2049*4*4
  int*  maxbuf = (int*)(Rbuf + (size_t)N_WIN * N_BINS * N_SRC * 4);

  init_kernel<<<(N_FRAMES * 14 + 255) / 256, 256, 0, stream>>>(Ycf, maxbuf);
  gen_irfft_kernel<<<(GEMM_K * GEMM_N) / 256, 256, 0, stream>>>(G);
  maxabs_kernel<<<N_WIN * 128, 256, 0, stream>>>(mix, maxbuf);
  wiener_R_kernel<<<(N_WIN * N_BINS * N_SRC + 255) / 256, 256, 0, stream>>>(
      spec, mix, maxbuf, Rbuf);
  wiener_apply_kernel<<<(N_WIN * WIN_T * N_BINS + 255) / 256, 256, 0, stream>>>(
      spec, mix, Rbuf, maxbuf, Ycf);
  wmma_gemm_kernel<<<(GEMM_M / 64) * (GEMM_N / 64), 128, 0, stream>>>(Ycf, G, Cmat);
  ola_kernel<<<(16 * AUDIO_LEN + 255) / 256, 256, 0, stream>>>(Cmat, out);
}